// QuantumLayer_82377472737504
// MI455X (gfx1250) — compile-verified
//
#include <hip/hip_runtime.h>

// ---------------------------------------------------------------------------
// 12-qubit / 4-layer RX + ring-CNOT variational circuit on MI455X (gfx1250).
//
// Roofline: 8192 statevectors x 32KB live entirely in LDS; global I/O is only
// ~0.8 MB (~35ns @ 23.3 TB/s) so the kernel is pure compute. One workgroup
// (8 x wave32) per batch element, ping-pong state buffers in LDS (2 x 32KB).
//
// Key trick (this revision): work in the PHASE-TWISTED basis
//   s~[i] = i^popcount(i) * s[i].
// Every fused 4-qubit RX group unitary G (entries mag * (-i)^pc(r^c)) becomes
// REAL there: G'[r,c] = mag * (-1)^pc(c & ~r). So each complex gate apply is
// 2 real WMMA matmuls instead of 4, and the initial product state is purely
// real -> layer 0 runs real-only (half again). The ring-CNOT entangler in the
// twisted basis is the closed-form bit map
//   b_p' = b_p^b_{p+1} (p=0..9), b11' = b11^b0, b10' = b10^b11^b0
// times a per-element phase i^{pc(i)-pc(p(i))} in {1,i,-1,-i}, fused into the
// next layer's TOP gather as a re/im swap-negate. Readout (|amp|^2) is phase
// invariant -> final entangler fuses into the readout gather with no fixup.
//
// CDNA5 mapping:
//  * gates: V_WMMA_F32_16X16X4_F32, K=16 as 4 chained WMMAs; G' fragments
//    built per-lane in registers (zero memory traffic for gate matrices).
//  * trig: hardware v_sin_f32/v_cos_f32 (__sinf/__cosf), args are O(1).
//  * <Z_k> readout: qubits 0-7 signs are per-thread-constant; wave32 shfl
//    reduction + LDS combine.
//
// WMMA layouts per CDNA5 ISA: A(16x4 f32) v2f/lane (lane<16:
// K={4s,4s+1}; lane>=16: K={4s+2,4s+3}); C/D v8f (VGPR v: M=v / v+8).
// B(4x16) assumed to mirror A's K split. Each group's 16 output tiles are
// disjoint 256-amplitude slices -> race-free in-place updates, barriers only
// between groups.
// ---------------------------------------------------------------------------

#define NQ 12
#define DIM 4096
#define NLAYERS 4
#define TPB 256
#define NWAVES 8
#define APT 16 /* amplitudes per thread */

typedef __attribute__((ext_vector_type(2))) float v2f;
typedef __attribute__((ext_vector_type(4))) float v4f;
typedef __attribute__((ext_vector_type(8))) float v8f;

__device__ __forceinline__ v8f wmma4(v2f a, v2f b, v8f c) {
  // D(16x16) = A(16x4)*B(4x16) + C on the fp32 matrix pipe
  return __builtin_amdgcn_wmma_f32_16x16x4_f32(false, a, false, b, (short)0, c,
                                               false, false);
}

// Closed form of the composed ring-CNOT permutation p (new[i] = old[p(i)]).
__device__ __forceinline__ int ringperm(int i) {
  int x11 = ((i >> 11) ^ i) & 1;
  return ((i ^ (i >> 1)) & 0x3FF) | ((((i >> 10) ^ x11) & 1) << 10) |
         (x11 << 11);
}

// Twisted-basis group entry: G'[r,c] = prod_j (r_j==c_j ? cos : sin)
//                                      * (-1)^popcount(c & ~r), real.
__device__ __forceinline__ float gq_entry(const float* cs, const float* sn,
                                          int r, int c) {
  float mag = 1.0f;
  int sgn = 0;
#pragma unroll
  for (int jj = 0; jj < 4; ++jj) {
    int rb = (r >> (3 - jj)) & 1;
    int cb = (c >> (3 - jj)) & 1;
    mag *= (rb == cb) ? cs[jj] : sn[jj];
    sgn += cb & ~rb;
  }
  return (sgn & 1) ? -mag : mag;
}

// Apply one fused 4-qubit group unitary (real in the twisted basis).
// G==0 (contraction on top nibble) reads src, writes dst; for REAL==false it
// also applies the fused entangler map + i^phi phase fixup on the gather.
// G==1/2 run in place on dst. REAL==true: state imaginary part is zero
// (layer 0); TOP writes the zeros, MID/BOT leave them untouched.
template <int G, bool REAL>
__device__ __forceinline__ void apply_group(const float* c4, const float* s4,
                                            const float* srcRe,
                                            const float* srcIm, float* dstRe,
                                            float* dstIm, int wave, int hi,
                                            int ln) {
  // Per-lane gate fragments: A = G' for G<2 (D = G'*S), B = G'^T for G==2
  // (D = S*G'^T); same per-lane formula either way.
  v2f fR[4];
#pragma unroll
  for (int s = 0; s < 4; ++s) {
    int c0 = 4 * s + 2 * hi;
    fR[s] = {gq_entry(c4, s4, ln, c0), gq_entry(c4, s4, ln, c0 + 1)};
  }

#pragma unroll
  for (int tt = 0; tt < 2; ++tt) {
    int t = wave + tt * NWAVES; // tile 0..15, disjoint 256-amp slices

    v2f bRe[4], bIm[4];
#pragma unroll
    for (int s = 0; s < 4; ++s) {
      int k0 = 4 * s + 2 * hi;
      if (G == 0) {
        int ia = (k0 << 8) | (16 * t + ln); // B[k=a][nn], cols 16t+nn
        int ib = ia + 256;
        if (REAL) { // layer 0: identity gather, real state
          bRe[s] = {srcRe[ia], srcRe[ib]};
        } else { // fused entangler: gather at p(i), multiply by i^phi
          int qa = ringperm(ia), qb = ringperm(ib);
          float ra = srcRe[qa], ma = srcIm[qa];
          float rb = srcRe[qb], mb = srcIm[qb];
          int fa = (__popc(ia) - __popc(qa)) & 3;
          int fb = (__popc(ib) - __popc(qb)) & 3;
          float xra = (fa & 1) ? ((fa & 2) ? ma : -ma) : ((fa & 2) ? -ra : ra);
          float xia = (fa & 1) ? ((fa & 2) ? -ra : ra) : ((fa & 2) ? -ma : ma);
          float xrb = (fb & 1) ? ((fb & 2) ? mb : -mb) : ((fb & 2) ? -rb : rb);
          float xib = (fb & 1) ? ((fb & 2) ? -rb : rb) : ((fb & 2) ? -mb : mb);
          bRe[s] = {xra, xrb};
          bIm[s] = {xia, xib};
        }
      } else if (G == 1) { // tile a=t; B[k=m'][nn=r]
        int ia = (t << 8) | (k0 << 4) | ln;
        bRe[s] = {srcRe[ia], srcRe[ia + 16]};
        if (!REAL) bIm[s] = {srcIm[ia], srcIm[ia + 16]};
      } else { // A = state rows (a,m); col = r' -> contiguous pair, b64 load
        int ia = ((16 * t + ln) << 4) | k0; // k0 even -> 8B aligned
        bRe[s] = *reinterpret_cast<const v2f*>(&srcRe[ia]);
        if (!REAL) bIm[s] = *reinterpret_cast<const v2f*>(&srcIm[ia]);
      }
    }

    // Real gate matrix: DRe = G'*SRe, DIm = G'*SIm (or S*G'^T for G==2).
    v8f dR = {};
    v8f dI = {};
    if (G < 2) {
#pragma unroll
      for (int s = 0; s < 4; ++s) dR = wmma4(fR[s], bRe[s], dR);
      if (!REAL) {
#pragma unroll
        for (int s = 0; s < 4; ++s) dI = wmma4(fR[s], bIm[s], dI);
      }
    } else {
#pragma unroll
      for (int s = 0; s < 4; ++s) dR = wmma4(bRe[s], fR[s], dR);
      if (!REAL) {
#pragma unroll
        for (int s = 0; s < 4; ++s) dI = wmma4(bIm[s], fR[s], dI);
      }
    }

    // C/D layout: VGPR v holds M=v (lanes 0-15) / M=v+8 (lanes 16-31)
#pragma unroll
    for (int v = 0; v < 8; ++v) {
      int lrow = v + 8 * hi;
      int i;
      if (G == 0)
        i = (lrow << 8) | (16 * t + ln);
      else if (G == 1)
        i = (t << 8) | (lrow << 4) | ln;
      else
        i = ((16 * t + lrow) << 4) | ln;
      dstRe[i] = dR[v];
      if (!REAL)
        dstIm[i] = dI[v];
      else if (G == 0)
        dstIm[i] = 0.0f; // establish Im=0 once; MID/BOT keep it
    }
  }
}

__global__ __launch_bounds__(TPB) void qlayer_kernel(
    const float* __restrict__ x, const float* __restrict__ w,
    float* __restrict__ out) {
  __shared__ float bufRe[2][DIM];
  __shared__ float bufIm[2][DIM];
  __shared__ float wpart[NWAVES][NQ];

  const int b = blockIdx.x;
  const int tid = threadIdx.x;
  const int lane = tid & 31;
  const int wave = tid >> 5;
  const int hi = lane >> 4; // 0: lanes 0-15, 1: lanes 16-31
  const int ln = lane & 15;

  // ------------- Phase 1: product state; REAL in the twisted basis ---------
  {
    float cs[NQ], sn[NQ];
#pragma unroll
    for (int j = 0; j < NQ; ++j) {
      float h = 0.5f * x[b * NQ + j]; // uniform per block -> scalar loads
      cs[j] = __cosf(h);              // v_cos_f32: args O(1), no slow path
      sn[j] = __sinf(h);              // v_sin_f32
    }
    // i = tid*16 + k: bits 11..4 of i = tid (qubits 0-7), bits 3..0 = k.
    float hiMag = 1.0f;
#pragma unroll
    for (int j = 0; j < 8; ++j)
      hiMag *= ((tid >> (7 - j)) & 1) ? sn[j] : cs[j];
    // 16 possible low products as a 2x2 tree (qubit pairs {8,9} and {10,11})
    float pA[4] = {cs[8] * cs[9], cs[8] * sn[9], sn[8] * cs[9], sn[8] * sn[9]};
    float pB[4] = {cs[10] * cs[11], cs[10] * sn[11], sn[10] * cs[11],
                   sn[10] * sn[11]};
    v4f vr[4];
#pragma unroll
    for (int k = 0; k < APT; ++k) // i^pc * (-i)^pc = 1: amplitude is just mag
      vr[k >> 2][k & 3] = hiMag * pA[k >> 2] * pB[k & 3];
#pragma unroll
    for (int m = 0; m < 4; ++m) // contiguous, 16B aligned -> ds_store_b128
      *reinterpret_cast<v4f*>(&bufRe[0][tid * APT + 4 * m]) = vr[m];
  }
  __syncthreads();

  // ------------- Phase 2: weight layers ------------------------------------
  float cw[NQ], sw[NQ];
#pragma unroll
  for (int j = 0; j < NQ; ++j) {
    float h = 0.5f * w[j]; // layer 0 weights
    cw[j] = __cosf(h);
    sw[j] = __sinf(h);
  }
  // Layer 0: state is purely real in the twisted basis -> real-only path.
  apply_group<0, true>(cw + 0, sw + 0, bufRe[0], bufIm[0], bufRe[1], bufIm[1],
                       wave, hi, ln);
  __syncthreads();
  apply_group<1, true>(cw + 4, sw + 4, bufRe[1], bufIm[1], bufRe[1], bufIm[1],
                       wave, hi, ln);
  __syncthreads();
  apply_group<2, true>(cw + 8, sw + 8, bufRe[1], bufIm[1], bufRe[1], bufIm[1],
                       wave, hi, ln);
  __syncthreads();

  int cur = 1;
  for (int l = 1; l < NLAYERS; ++l) {
#pragma unroll
    for (int j = 0; j < NQ; ++j) {
      float h = 0.5f * w[l * NQ + j];
      cw[j] = __cosf(h);
      sw[j] = __sinf(h);
    }
    float* sR = bufRe[cur];
    float* sI = bufIm[cur];
    float* dR = bufRe[cur ^ 1];
    float* dI = bufIm[cur ^ 1];

    apply_group<0, false>(cw + 0, sw + 0, sR, sI, dR, dI, wave, hi, ln);
    __syncthreads();
    apply_group<1, false>(cw + 4, sw + 4, dR, dI, dR, dI, wave, hi, ln);
    __syncthreads();
    apply_group<2, false>(cw + 8, sw + 8, dR, dI, dR, dI, wave, hi, ln);
    __syncthreads();
    cur ^= 1;
  }

  // ------------- Phase 3: <Z_k> readout (final entangler fused) ------------
  // |amp|^2 is invariant under the phase twist -> no fixup needed.
  {
    const float* fR = bufRe[cur];
    const float* fI = bufIm[cur];
    float totP = 0.0f, s8 = 0.0f, s9 = 0.0f, s10 = 0.0f, s11 = 0.0f;
#pragma unroll
    for (int k = 0; k < APT; ++k) {
      int q = ringperm(tid * APT + k); // fused last-layer entangler
      float re = fR[q], im = fI[q];
      float p = re * re + im * im;
      totP += p;
      s8 += (k & 8) ? -p : p;  // qubit 8  <-> bit 3 of k
      s9 += (k & 4) ? -p : p;  // qubit 9  <-> bit 2
      s10 += (k & 2) ? -p : p; // qubit 10 <-> bit 1
      s11 += (k & 1) ? -p : p; // qubit 11 <-> bit 0
    }
    float acc[NQ];
#pragma unroll
    for (int q2 = 0; q2 < 8; ++q2) // qubits 0-7: sign constant per thread
      acc[q2] = ((tid >> (7 - q2)) & 1) ? -totP : totP;
    acc[8] = s8;
    acc[9] = s9;
    acc[10] = s10;
    acc[11] = s11;
    // wave32 reduction
#pragma unroll
    for (int off = 16; off >= 1; off >>= 1) {
#pragma unroll
      for (int q2 = 0; q2 < NQ; ++q2) acc[q2] += __shfl_down(acc[q2], off);
    }
    if (lane == 0) {
#pragma unroll
      for (int q2 = 0; q2 < NQ; ++q2) wpart[wave][q2] = acc[q2];
    }
    __syncthreads();
    if (tid < NQ) {
      float ssum = 0.0f;
#pragma unroll
      for (int wv = 0; wv < NWAVES; ++wv) ssum += wpart[wv][tid];
      out[b * NQ + tid] = ssum;
    }
  }
}

extern "C" void kernel_launch(void* const* d_in, const int* in_sizes, int n_in,
                              void* d_out, int out_size, void* d_ws,
                              size_t ws_size, hipStream_t stream) {
  (void)n_in;
  (void)out_size;
  (void)d_ws;
  (void)ws_size;
  const float* x = (const float*)d_in[0]; // (64,128,12) fp32
  const float* w = (const float*)d_in[1]; // (4,12) fp32
  float* out = (float*)d_out;             // (64,128,12) fp32
  const int Btot = in_sizes[0] / NQ;      // 8192 statevectors
  qlayer_kernel<<<Btot, TPB, 0, stream>>>(x, w, out);
}